// FunnelGNN_44375602103091
// MI455X (gfx1250) — compile-verified
//
#include <hip/hip_runtime.h>
#include <float.h>
#include <math.h>

typedef float v2f __attribute__((ext_vector_type(2)));
typedef float v8f __attribute__((ext_vector_type(8)));

#define NNODES 65536
#define NEDGES 1048576
#define NBATCH 512
#define GCOLS  2304
#define BN_EPS 1e-5f
#define SLOPE  0.01f

// ---- order-preserving float <-> uint encoding for atomicMax on floats ----
__device__ __forceinline__ unsigned encf(float f) {
    unsigned u = __float_as_uint(f);
    return (u & 0x80000000u) ? ~u : (u | 0x80000000u);
}
__device__ __forceinline__ float decf(unsigned u) {
    return __uint_as_float((u & 0x80000000u) ? (u & 0x7FFFFFFFu) : ~u);
}

// -------------------- utility kernels --------------------
__global__ void fill_kernel(float* __restrict__ p, float v, long long n) {
    long long i = (long long)blockIdx.x * blockDim.x + threadIdx.x;
    if (i < n) p[i] = v;
}

__global__ void count_kernel(const int* __restrict__ batch, float* __restrict__ cnt) {
    int i = blockIdx.x * blockDim.x + threadIdx.x;
    if (i < NNODES) atomicAdd(&cnt[batch[i]], 1.0f);
}

__global__ void inv_kernel(const float* __restrict__ cnt, float* __restrict__ inv) {
    int b = blockIdx.x * blockDim.x + threadIdx.x;
    if (b < NBATCH) inv[b] = 1.0f / fmaxf(cnt[b], 1.0f);
}

// g layout per row (2304): [L1: max128|mean128|sum128][L2: max256|mean256|sum256][L3: max384|mean384|sum384]
__device__ __forceinline__ void g_col_info(int c, int& col0, int& Nd) {
    if (c < 384)       { col0 = 0;    Nd = 128; }
    else if (c < 1152) { col0 = 384;  Nd = 256; }
    else               { col0 = 1152; Nd = 384; }
}

__global__ void init_g_kernel(float* __restrict__ g) {
    int idx = blockIdx.x * blockDim.x + threadIdx.x;
    if (idx >= NBATCH * GCOLS) return;
    int c = idx % GCOLS, col0, Nd;
    g_col_info(c, col0, Nd);
    if (c - col0 < Nd) ((unsigned*)g)[idx] = encf(-FLT_MAX);  // max slots
    else               g[idx] = 0.0f;                          // mean/sum slots
}

__global__ void finalize_pool_kernel(float* __restrict__ g, const float* __restrict__ inv) {
    int idx = blockIdx.x * blockDim.x + threadIdx.x;
    if (idx >= NBATCH * GCOLS) return;
    int b = idx / GCOLS, c = idx % GCOLS, col0, Nd;
    g_col_info(c, col0, Nd);
    int local = c - col0;
    if (local < Nd) {
        g[idx] = decf(((const unsigned*)g)[idx]);                       // decode max
    } else if (local < 2 * Nd) {
        g[idx] = g[(long long)b * GCOLS + col0 + local + Nd] * inv[b];  // mean = sum * inv_cnt
    }                                                                   // sum slot stays
}

// -------------------- edge scatter-add (memory-bound phase) --------------------
__global__ void aggregate_kernel(const float* __restrict__ x, const int* __restrict__ ei,
                                 float* __restrict__ agg, int d) {
    long long tid = (long long)blockIdx.x * blockDim.x + threadIdx.x;
    int per = d >> 2;                       // threads per edge (float4 per thread)
    long long e = tid / per;
    int c = (int)(tid % per) << 2;
    if (e >= NEDGES) return;
    int src = ei[e];                        // edge_index[0][e]
    int dst = ei[NEDGES + e];               // edge_index[1][e]
    const float4 v = *(const float4*)(x + (size_t)src * d + c);
    float* p = agg + (size_t)dst * d + c;
    atomicAdd(p + 0, v.x); atomicAdd(p + 1, v.y);
    atomicAdd(p + 2, v.z); atomicAdd(p + 3, v.w);
}

// -------------------- fused WMMA GraphConv + bias + LeakyReLU + BN + pooling --------------------
// out = BN(lrelu(agg@Wrel + brel + x@Wroot)); pooled (max/sum) into gpool via atomics.
// One wave computes a 16(M) x 64(N) tile with V_WMMA_F32_16X16X4_F32.
// Both weight tiles (K x 64 each) staged in LDS once per block; padded stride 72
// floats keeps ds_load B-fragment reads bank-conflict free and B128-store aligned.
template <int K>
__global__ __launch_bounds__(128)
void gconv_wmma_kernel(const float* __restrict__ agg, const float* __restrict__ xin,
                       const float* __restrict__ Wrel, const float* __restrict__ brel,
                       const float* __restrict__ Wroot,
                       const float* __restrict__ bng, const float* __restrict__ bnb,
                       const float* __restrict__ bnm, const float* __restrict__ bnv,
                       const int* __restrict__ batch,
                       float* __restrict__ hout,      // may be nullptr (last layer)
                       float* __restrict__ gpool,     // [NBATCH, GCOLS]
                       int Nd, int col0) {
    constexpr int WST = 72;                 // LDS row stride (floats)
    __shared__ float wlds[2 * K * WST];
    float* wrel_s  = wlds;
    float* wroot_s = wlds + K * WST;

    const int lane = threadIdx.x & 31;
    const int wave = threadIdx.x >> 5;
    const int r    = lane & 15;             // M (for A) / N (for B,C,D) index within tile
    const int half = lane >> 4;             // selects K-pair (A,B) / M+8 (C,D)
    const int row0 = blockIdx.y * 64 + wave * 16;
    const int nb   = blockIdx.x * 64;

    // ---- stage both weight tiles (K x 64) into LDS, coalesced float4 ----
    {
        const int lrow = threadIdx.x >> 4;        // 0..7
        const int lcol = (threadIdx.x & 15) * 4;  // 0..60
#pragma unroll 2
        for (int rb = 0; rb < K; rb += 8) {
            int row = rb + lrow;
            float4 vr = *(const float4*)(Wrel  + (size_t)row * Nd + nb + lcol);
            float4 vt = *(const float4*)(Wroot + (size_t)row * Nd + nb + lcol);
            *(float4*)&wrel_s [row * WST + lcol] = vr;
            *(float4*)&wroot_s[row * WST + lcol] = vt;
        }
    }
    __syncthreads();

    v8f acc[4] = {};
    const float* arow_agg = agg + (size_t)(row0 + r) * K;
    const float* arow_x   = xin + (size_t)(row0 + r) * K;

    for (int kb = 0; kb < K; kb += 16) {
        // stream-ahead prefetch of the A rows (speculative OOB prefetch is dropped)
        __builtin_prefetch(arow_agg + kb + 64, 0, 1);
        __builtin_prefetch(arow_x   + kb + 64, 0, 1);
#pragma unroll
        for (int k = kb; k < kb + 16; k += 4) {
            // A frags: lane holds A[r, k+2*half], A[r, k+2*half+1]  (ISA 16x4 f32 layout)
            v2f a_agg = *(const v2f*)(arow_agg + k + 2 * half);
            v2f a_x   = *(const v2f*)(arow_x   + k + 2 * half);
            const float* wr = &wrel_s [(k + 2 * half) * WST];
            const float* wt = &wroot_s[(k + 2 * half) * WST];
#pragma unroll
            for (int j = 0; j < 4; ++j) {
                int jc = j * 16 + r;
                // B frags: VGPR v holds rows K = v + 2*half, cols = lane&15
                v2f b_rel = { wr[jc], wr[WST + jc] };
                v2f b_rt  = { wt[jc], wt[WST + jc] };
                acc[j] = __builtin_amdgcn_wmma_f32_16x16x4_f32(
                    false, a_agg, false, b_rel, (short)0, acc[j], false, false);
                acc[j] = __builtin_amdgcn_wmma_f32_16x16x4_f32(
                    false, a_x,   false, b_rt,  (short)0, acc[j], false, false);
            }
        }
    }

    // epilogue: bias -> lrelu -> BN -> (store h) + pooled atomics
#pragma unroll
    for (int j = 0; j < 4; ++j) {
        int col = nb + j * 16 + r;
        float bias = brel[col];
        float sc = bng[col] * rsqrtf(bnv[col] + BN_EPS);
        float sh = bnb[col] - bnm[col] * sc;
#pragma unroll
        for (int v = 0; v < 8; ++v) {
            int row = row0 + v + 8 * half;   // C/D layout: VGPR v holds rows v and v+8
            float val = acc[j][v] + bias;
            val = (val >= 0.0f) ? val : SLOPE * val;
            val = val * sc + sh;
            if (hout) hout[(size_t)row * Nd + col] = val;
            int b = batch[row];
            float* gb = gpool + (size_t)b * GCOLS + col0;
            atomicMax((unsigned*)(gb + col), encf(val));     // segment max
            atomicAdd(gb + 2 * Nd + col, val);               // segment sum (mean derived later)
        }
    }
}

// -------------------- FC head --------------------
__global__ void fc1_kernel(const float* __restrict__ g, const float* __restrict__ W,
                           const float* __restrict__ bias, float* __restrict__ out) {
    int idx = blockIdx.x * blockDim.x + threadIdx.x;
    if (idx >= NBATCH * 128) return;
    int b = idx >> 7, o = idx & 127;
    const float* gr = g + (size_t)b * GCOLS;
    float s = bias[o];
    for (int k = 0; k < GCOLS; ++k) s = fmaf(gr[k], W[(size_t)k * 128 + o], s);
    out[idx] = s;
}

__global__ void fc2_lsm_kernel(const float* __restrict__ h, const float* __restrict__ W,
                               const float* __restrict__ bias, float* __restrict__ out) {
    int b = blockIdx.x * blockDim.x + threadIdx.x;
    if (b >= NBATCH) return;
    const float* hr = h + (size_t)b * 128;
    float z0 = bias[0], z1 = bias[1];
    for (int k = 0; k < 128; ++k) {
        float v = hr[k];
        z0 = fmaf(v, W[k * 2 + 0], z0);
        z1 = fmaf(v, W[k * 2 + 1], z1);
    }
    float mx  = fmaxf(z0, z1);
    float lse = mx + logf(expf(z0 - mx) + expf(z1 - mx));
    out[b * 2 + 0] = z0 - lse;
    out[b * 2 + 1] = z1 - lse;
}

// -------------------- launch --------------------
static inline void fill_async(float* p, float v, long long n, hipStream_t s) {
    fill_kernel<<<(unsigned)((n + 255) / 256), 256, 0, s>>>(p, v, n);
}

extern "C" void kernel_launch(void* const* d_in, const int* in_sizes, int n_in,
                              void* d_out, int out_size, void* d_ws, size_t ws_size,
                              hipStream_t stream) {
    const float* x     = (const float*)d_in[0];
    const int*   ei    = (const int*)d_in[1];
    const int*   batch = (const int*)d_in[2];
    // per-layer params: W_rel, b_rel, W_root, bn_g, bn_b, bn_m, bn_v
    const float* P[21];
    for (int i = 0; i < 21; ++i) P[i] = (const float*)d_in[3 + i];
    const float* lin1_W = (const float*)d_in[24];
    const float* lin1_b = (const float*)d_in[25];
    const float* lin2_W = (const float*)d_in[26];
    const float* lin2_b = (const float*)d_in[27];
    float* out = (float*)d_out;

    float* ws   = (float*)d_ws;
    float* h1   = ws;                                   // N*128
    float* h2   = h1 + (size_t)NNODES * 128;            // N*256
    float* agg  = h2 + (size_t)NNODES * 256;            // N*256 (reused per layer)
    float* g    = agg + (size_t)NNODES * 256;           // B*2304
    float* cnt  = g + (size_t)NBATCH * GCOLS;           // B
    float* inv  = cnt + NBATCH;                         // B
    float* fc1o = inv + NBATCH;                         // B*128

    // graph sizes / counts
    fill_async(cnt, 0.0f, NBATCH, stream);
    count_kernel<<<NNODES / 256, 256, 0, stream>>>(batch, cnt);
    inv_kernel<<<(NBATCH + 255) / 256, 256, 0, stream>>>(cnt, inv);
    init_g_kernel<<<(NBATCH * GCOLS + 255) / 256, 256, 0, stream>>>(g);

    // ---- layer 1: x(128) -> h1(128) ----
    fill_async(agg, 0.0f, (long long)NNODES * 128, stream);
    aggregate_kernel<<<(unsigned)(((long long)NEDGES * 32 + 255) / 256), 256, 0, stream>>>(x, ei, agg, 128);
    gconv_wmma_kernel<128><<<dim3(128 / 64, NNODES / 64), 128, 0, stream>>>(
        agg, x, P[0], P[1], P[2], P[3], P[4], P[5], P[6], batch, h1, g, 128, 0);

    // ---- layer 2: h1(128) -> h2(256) ----
    fill_async(agg, 0.0f, (long long)NNODES * 128, stream);
    aggregate_kernel<<<(unsigned)(((long long)NEDGES * 32 + 255) / 256), 256, 0, stream>>>(h1, ei, agg, 128);
    gconv_wmma_kernel<128><<<dim3(256 / 64, NNODES / 64), 128, 0, stream>>>(
        agg, h1, P[7], P[8], P[9], P[10], P[11], P[12], P[13], batch, h2, g, 256, 384);

    // ---- layer 3: h2(256) -> pooled only (h3 never materialized) ----
    fill_async(agg, 0.0f, (long long)NNODES * 256, stream);
    aggregate_kernel<<<(unsigned)(((long long)NEDGES * 64 + 255) / 256), 256, 0, stream>>>(h2, ei, agg, 256);
    gconv_wmma_kernel<256><<<dim3(384 / 64, NNODES / 64), 128, 0, stream>>>(
        agg, h2, P[14], P[15], P[16], P[17], P[18], P[19], P[20], batch, nullptr, g, 384, 1152);

    // ---- pooling finalize + FC head ----
    finalize_pool_kernel<<<(NBATCH * GCOLS + 255) / 256, 256, 0, stream>>>(g, inv);
    fc1_kernel<<<(NBATCH * 128 + 255) / 256, 256, 0, stream>>>(g, lin1_W, lin1_b, fc1o);
    fc2_lsm_kernel<<<(NBATCH + 127) / 128, 128, 0, stream>>>(fc1o, lin2_W, lin2_b, out);
}